// Encoder_Gnn_82721070121623
// MI455X (gfx1250) — compile-verified
//
#include <hip/hip_runtime.h>
#include <stddef.h>

typedef float v2f __attribute__((ext_vector_type(2)));
typedef float v8f __attribute__((ext_vector_type(8)));

#define SEQL 1024
#define HD   1024
#define G4   4096
#define DIRSTRIDE 4194304ul   // 1024*4096 floats

// ---------------------------------------------------------------- bias sum
__global__ void bias_sum_kernel(const float* __restrict__ a,
                                const float* __restrict__ b,
                                float* __restrict__ o) {
    int i = blockIdx.x * 256 + threadIdx.x;   // 4096 total
    if (i < G4) o[i] = a[i] + b[i];
}

// ------------------------------------------------- Wih transpose: WT[k][r]
__global__ void tr_wih_kernel(const float* __restrict__ W,  // [4096][K]
                              float* __restrict__ WT,       // [K][4096]
                              int K) {
    size_t tid = (size_t)blockIdx.x * 256 + threadIdx.x;
    size_t total = (size_t)K * G4;
    if (tid >= total) return;
    int r = (int)(tid & 4095);
    int k = (int)(tid >> 12);
    WT[tid] = W[(size_t)r * K + k];
}

// ------------------------- Whh transpose+interleave: WT[k][j][gate] float4
__global__ void tr_whh_kernel(const float* __restrict__ W,  // [4*1024][1024]
                              float* __restrict__ WT) {     // [1024][1024][4]
    size_t tid = (size_t)blockIdx.x * 256 + threadIdx.x;    // 4M total
    if (tid >= DIRSTRIDE) return;
    int g = (int)(tid & 3);
    size_t r = tid >> 2;
    int j = (int)(r & 1023);
    int k = (int)(r >> 10);
    WT[tid] = W[((size_t)g * HD + j) * HD + k];
}

// --------------------------------------- WMMA fp32 GEMM: Z = A * WT + bias
// Each wave computes a 16x64 strip (4 accumulators, A fragment reused 4x).
// A: [1024][K] per dir (optionally row-reversed for dir1 = backward, layer0)
// WT: [2][K][4096]  bias: [2][4096]  Z: [2][1024][4096]
__global__ __launch_bounds__(128)
void gemm_z_kernel(const float* __restrict__ Ad0,
                   const float* __restrict__ Ad1,
                   const float* __restrict__ WT,
                   const float* __restrict__ bias,
                   float* __restrict__ Z,
                   int K, int revDir1) {
    const int wave = threadIdx.x >> 5;
    const int lane = threadIdx.x & 31;
    const int dir  = blockIdx.y;
    const int tile = blockIdx.x * 4 + wave;      // 0..4095
    const int tm   = (tile >> 6) << 4;           // seq tile offset (64 M tiles)
    const int tng  = (tile & 63) << 6;           // gate strip offset (64 strips of 64)

    const float* A = dir ? Ad1 : Ad0;
    const float* W = WT + (size_t)dir * DIRSTRIDE;

    const int m  = lane & 15;
    const int kk = (lane >> 4) << 1;             // 0 or 2
    int srow = tm + m;
    if (dir && revDir1) srow = SEQL - 1 - srow;
    const float* arow = A + (size_t)srow * K;
    const int n = lane & 15;

    v8f c[4] = {{}, {}, {}, {}};
    for (int k0 = 0; k0 < K; k0 += 4) {
        v2f a;
        a.x = arow[k0 + kk];
        a.y = arow[k0 + kk + 1];
        const float* wp = W + (size_t)(k0 + kk) * G4 + tng + n;
        if (k0 + 32 < K)
            __builtin_prefetch(W + (size_t)(k0 + 32 + kk) * G4 + tng + n, 0, 1);
        v2f b0, b1, b2, b3;
        b0.x = wp[0];   b0.y = wp[G4];
        b1.x = wp[16];  b1.y = wp[G4 + 16];
        b2.x = wp[32];  b2.y = wp[G4 + 32];
        b3.x = wp[48];  b3.y = wp[G4 + 48];
        c[0] = __builtin_amdgcn_wmma_f32_16x16x4_f32(false, a, false, b0,
                                                     (short)0, c[0], false, false);
        c[1] = __builtin_amdgcn_wmma_f32_16x16x4_f32(false, a, false, b1,
                                                     (short)0, c[1], false, false);
        c[2] = __builtin_amdgcn_wmma_f32_16x16x4_f32(false, a, false, b2,
                                                     (short)0, c[2], false, false);
        c[3] = __builtin_amdgcn_wmma_f32_16x16x4_f32(false, a, false, b3,
                                                     (short)0, c[3], false, false);
    }
    const int mo = (lane >> 4) << 3;             // 0 or 8
    float* zb = Z + ((size_t)dir * SEQL + tm + mo) * G4 + tng + n;
    const float* bp = bias + (size_t)dir * G4 + tng + n;
#pragma unroll
    for (int u = 0; u < 4; ++u) {
        const float bv = bp[u * 16];
#pragma unroll
        for (int r = 0; r < 8; ++r)
            zb[(size_t)r * G4 + u * 16] = c[u][r] + bv;
    }
}

// -------------------------------------------------------------- activations
__device__ __forceinline__ float sigm(float x) {
    return 1.0f / (1.0f + __expf(-x));
}

__device__ __forceinline__ float gload(const float* p) {
    unsigned u = __hip_atomic_load((const unsigned*)p, __ATOMIC_RELAXED,
                                   __HIP_MEMORY_SCOPE_AGENT);
    return __uint_as_float(u);
}
__device__ __forceinline__ void gstore(float* p, float v) {
    __hip_atomic_store((unsigned*)p, __float_as_uint(v), __ATOMIC_RELAXED,
                       __HIP_MEMORY_SCOPE_AGENT);
}

// ----------------------------------------------- persistent recurrent scan
// grid: (16 blocks, 2 dirs) x 256 threads. thread = (unit j in block, K-quarter q).
__global__ __launch_bounds__(256)
void lstm_scan_kernel(const float* __restrict__ Z,      // [2][1024][4096]
                      const float* __restrict__ WhhTI,  // [2][1024][1024][4]
                      float* __restrict__ H0,           // [2][1024][1024] (layer0 out)
                      float* __restrict__ out,          // d_out base
                      float* __restrict__ hbuf,         // [2 dirs][2 phases][1024]
                      unsigned* __restrict__ cnt,       // [2 layers][2 dirs]
                      int layer) {
    const int dir = blockIdx.y;
    const int blk = blockIdx.x;          // 0..15
    const int tid = threadIdx.x;         // 0..255
    const int jl  = tid & 63;
    const int q   = tid >> 6;            // K-quarter
    const int j   = blk * 64 + jl;       // hidden unit 0..1023

    const float* Zd = Z + (size_t)dir * DIRSTRIDE;
    const float* Wd = WhhTI + (size_t)dir * DIRSTRIDE;
    float* hb[2] = { hbuf + (dir * 2 + 0) * HD, hbuf + (dir * 2 + 1) * HD };
    unsigned* bar = cnt + (layer * 2 + dir);

    __shared__ float sh[HD];
    __shared__ float red[256][4];

    float cst = 0.0f;                    // cell state (valid in q==0 threads)

    for (int t = 0; t < SEQL; ++t) {
        // stage h_{t-1} into LDS (L0-bypassing loads; t==0 -> zeros)
        const float* hp = hb[t & 1];
        for (int k = tid; k < HD; k += 256)
            sh[k] = (t == 0) ? 0.0f : gload(&hp[k]);
        __syncthreads();

        // partial dot products over this thread's K-quarter
        float pi = 0.f, pf = 0.f, pg = 0.f, po = 0.f;
        const float* wrow = Wd + (size_t)(q * 256) * (HD * 4) + (size_t)j * 4;
        const float* shq = sh + q * 256;
#pragma unroll 4
        for (int k = 0; k < 256; ++k) {
            const float hk = shq[k];
            const float4 w = *(const float4*)(wrow + (size_t)k * (HD * 4));
            pi = fmaf(w.x, hk, pi);
            pf = fmaf(w.y, hk, pf);
            pg = fmaf(w.z, hk, pg);
            po = fmaf(w.w, hk, po);
        }
        red[tid][0] = pi; red[tid][1] = pf; red[tid][2] = pg; red[tid][3] = po;
        __syncthreads();

        if (q == 0) {
            const float* zrow = Zd + (size_t)t * G4;
            float gi = zrow[0 * HD + j] + red[tid][0] + red[tid + 64][0] +
                       red[tid + 128][0] + red[tid + 192][0];
            float gf = zrow[1 * HD + j] + red[tid][1] + red[tid + 64][1] +
                       red[tid + 128][1] + red[tid + 192][1];
            float gg = zrow[2 * HD + j] + red[tid][2] + red[tid + 64][2] +
                       red[tid + 128][2] + red[tid + 192][2];
            float go = zrow[3 * HD + j] + red[tid][3] + red[tid + 64][3] +
                       red[tid + 128][3] + red[tid + 192][3];

            cst = sigm(gf) * cst + sigm(gi) * tanhf(gg);
            const float h = sigm(go) * tanhf(cst);

            gstore(&hb[(t & 1) ^ 1][j], h);
            if (layer == 0) {
                H0[((size_t)dir * SEQL + t) * HD + j] = h;
            } else {
                const int tt = dir ? (SEQL - 1 - t) : t;
                out[8192 + (size_t)tt * 2048 + dir * 1024 + j] = h;  // final_outputs
            }
            if (t == SEQL - 1) {
                out[layer * 2048 + dir * 1024 + j] = cst;            // cell_memories
                out[4096 + layer * 2048 + dir * 1024 + j] = h;       // hidden_states
            }
        }

        // dir-wide barrier across 16 blocks (monotonic counter)
        __threadfence();
        __syncthreads();
        if (tid == 0) {
            __hip_atomic_fetch_add(bar, 1u, __ATOMIC_RELEASE,
                                   __HIP_MEMORY_SCOPE_AGENT);
            const unsigned target = 16u * (unsigned)(t + 1);
            while (__hip_atomic_load(bar, __ATOMIC_ACQUIRE,
                                     __HIP_MEMORY_SCOPE_AGENT) < target)
                __builtin_amdgcn_s_sleep(1);
        }
        __syncthreads();
    }
}

// ---------------------------------------------------------------- launcher
extern "C" void kernel_launch(void* const* d_in, const int* in_sizes, int n_in,
                              void* d_out, int out_size, void* d_ws, size_t ws_size,
                              hipStream_t stream) {
    const float* x = (const float*)d_in[0];
    // input order: fw0 Wih,Whh,bih,bhh (1..4); fw1 (5..8); bw0 (9..12); bw1 (13..16)
    const float* fw0_Wih = (const float*)d_in[1];
    const float* fw0_Whh = (const float*)d_in[2];
    const float* fw0_bih = (const float*)d_in[3];
    const float* fw0_bhh = (const float*)d_in[4];
    const float* fw1_Wih = (const float*)d_in[5];
    const float* fw1_Whh = (const float*)d_in[6];
    const float* fw1_bih = (const float*)d_in[7];
    const float* fw1_bhh = (const float*)d_in[8];
    const float* bw0_Wih = (const float*)d_in[9];
    const float* bw0_Whh = (const float*)d_in[10];
    const float* bw0_bih = (const float*)d_in[11];
    const float* bw0_bhh = (const float*)d_in[12];
    const float* bw1_Wih = (const float*)d_in[13];
    const float* bw1_Whh = (const float*)d_in[14];
    const float* bw1_bih = (const float*)d_in[15];
    const float* bw1_bhh = (const float*)d_in[16];

    float* ws    = (float*)d_ws;
    float* Z     = ws;                         // 2*1024*4096
    float* WhhT  = Z    + 2 * DIRSTRIDE;       // 2*1024*1024*4
    float* WihT  = WhhT + 2 * DIRSTRIDE;       // 2*1024*4096 (layer K<=1024)
    float* H0    = WihT + 2 * DIRSTRIDE;       // 2*1024*1024
    float* bias  = H0   + 2ul * SEQL * HD;     // 2 layers * 2 dirs * 4096
    float* hbuf  = bias + 4ul * G4;            // 2*2*1024
    unsigned* cnt = (unsigned*)(hbuf + 4ul * HD);

    hipMemsetAsync(cnt, 0, 4 * sizeof(unsigned), stream);

    // bias sums: [layer][dir][4096]
    bias_sum_kernel<<<16, 256, 0, stream>>>(fw0_bih, fw0_bhh, bias + 0 * G4);
    bias_sum_kernel<<<16, 256, 0, stream>>>(bw0_bih, bw0_bhh, bias + 1 * G4);
    bias_sum_kernel<<<16, 256, 0, stream>>>(fw1_bih, fw1_bhh, bias + 2 * G4);
    bias_sum_kernel<<<16, 256, 0, stream>>>(bw1_bih, bw1_bhh, bias + 3 * G4);

    float* out = (float*)d_out;

    // ---------------- layer 0 ----------------
    tr_wih_kernel<<<(768 * G4 + 255) / 256, 256, 0, stream>>>(fw0_Wih, WihT, 768);
    tr_wih_kernel<<<(768 * G4 + 255) / 256, 256, 0, stream>>>(bw0_Wih, WihT + DIRSTRIDE, 768);
    tr_whh_kernel<<<(int)(DIRSTRIDE / 256), 256, 0, stream>>>(fw0_Whh, WhhT);
    tr_whh_kernel<<<(int)(DIRSTRIDE / 256), 256, 0, stream>>>(bw0_Whh, WhhT + DIRSTRIDE);
    gemm_z_kernel<<<dim3(1024, 2), 128, 0, stream>>>(x, x, WihT, bias + 0, Z, 768, 1);
    lstm_scan_kernel<<<dim3(16, 2), 256, 0, stream>>>(Z, WhhT, H0, out, hbuf, cnt, 0);

    // ---------------- layer 1 ----------------
    tr_wih_kernel<<<(int)((1024ul * G4) / 256), 256, 0, stream>>>(fw1_Wih, WihT, 1024);
    tr_wih_kernel<<<(int)((1024ul * G4) / 256), 256, 0, stream>>>(bw1_Wih, WihT + DIRSTRIDE, 1024);
    tr_whh_kernel<<<(int)(DIRSTRIDE / 256), 256, 0, stream>>>(fw1_Whh, WhhT);
    tr_whh_kernel<<<(int)(DIRSTRIDE / 256), 256, 0, stream>>>(bw1_Whh, WhhT + DIRSTRIDE);
    gemm_z_kernel<<<dim3(1024, 2), 128, 0, stream>>>(H0, H0 + (size_t)SEQL * HD,
                                                     WihT, bias + 2 * G4, Z, 1024, 0);
    lstm_scan_kernel<<<dim3(16, 2), 256, 0, stream>>>(Z, WhhT, H0, out, hbuf, cnt, 1);
}